// WavLMSelfAttention_9689446219991
// MI455X (gfx1250) — compile-verified
//
#include <hip/hip_runtime.h>
#include <hip/hip_bf16.h>
#include <math.h>

typedef __attribute__((ext_vector_type(2))) float v2f;
typedef __attribute__((ext_vector_type(8))) float v8f;
typedef __attribute__((ext_vector_type(4))) int v4i;

#define BSZ 8
#define LSEQ 1024
#define EMB 768
#define NH 12
#define DH 64
#define QKV_STRIDE (3 * EMB) // 2304

// LDS strides: multiples of 4 floats (16B, required for async b128 copies)
// AND bank-conflict-free for fragment reads (gcd-based lane->bank spread).
#define AS_STRIDE 36   // GEMM A/W tiles
#define KS_STRIDE 68   // attention K/V tiles
#define PS_STRIDE 18   // attention P relayout tile (VALU-written, b64 reads)

// ---------------------------------------------------------------------------
// CDNA5 async global->LDS copy path (ASYNCcnt), with synchronous fallback.
// Probe round 2 diagnostic: param0 of the b128 builtin is generic `v4i*`;
// LDS param follows the *3 (addrspace(3)) convention of global_load_lds.
// ---------------------------------------------------------------------------
#if defined(__has_builtin)
#if __has_builtin(__builtin_amdgcn_global_load_async_to_lds_b128)
#define HAVE_ASYNC_LDS 1
#endif
#endif
#ifndef HAVE_ASYNC_LDS
#define HAVE_ASYNC_LDS 0
#endif

#if HAVE_ASYNC_LDS
typedef __attribute__((address_space(3))) v4i* lds_v4i;
__device__ __forceinline__ void async_copy16(const float* g, float* l) {
    v4i* gp = (v4i*)g;                 // generic pointer, const dropped
    v4i* lp = (v4i*)l;                 // reinterpret pointee first
    lds_v4i l3 = (lds_v4i)lp;          // then addrspacecast generic -> LDS
    __builtin_amdgcn_global_load_async_to_lds_b128(gp, l3, 0, 0);
}
__device__ __forceinline__ void wait_async0() {
#if __has_builtin(__builtin_amdgcn_s_wait_asynccnt)
    __builtin_amdgcn_s_wait_asynccnt(0);
#else
    asm volatile("s_wait_asynccnt 0x0" ::: "memory");
#endif
}
#else
__device__ __forceinline__ void async_copy16(const float* g, float* l) {
    l[0] = g[0]; l[1] = g[1]; l[2] = g[2]; l[3] = g[3];
}
__device__ __forceinline__ void wait_async0() {}
#endif

// D = A(16x4) * B(4x16) + C, f32 WMMA (faithful to the f32 reference).
__device__ __forceinline__ v8f wmma_f32(v2f a, v2f b, v8f c) {
    return __builtin_amdgcn_wmma_f32_16x16x4_f32(
        false, a, false, b, (short)0, c, false, false);
}

// ---------------------------------------------------------------------------
// Kernel 1: position_bias_b (B,H,L,L) — T5 bucket per (q,k), gather embedding,
// broadcast over b,h with nontemporal (streaming) stores: 403MB, touched once.
// ---------------------------------------------------------------------------
__global__ __launch_bounds__(256) void bias_kernel(
    const float* __restrict__ emb, float* __restrict__ out2) {
    int gid = blockIdx.x * 256 + threadIdx.x;   // over L*L
    int q = gid >> 10;
    int k = gid & (LSEQ - 1);
    int rp = k - q;                             // mem - ctx
    int bucket = (rp > 0) ? 16 : 0;             // nb = 16
    int arp = rp < 0 ? -rp : rp;
    if (arp < 8) {
        bucket += arp;
    } else {
        int large = 8 + (int)(logf((float)arp * 0.125f) *
                              (8.0f / logf(16.0f)));
        bucket += (large < 15) ? large : 15;
    }
#pragma unroll
    for (int h = 0; h < NH; ++h) {
        float v = emb[bucket * NH + h];
#pragma unroll
        for (int b = 0; b < BSZ; ++b) {
            __builtin_nontemporal_store(
                v, &out2[(((size_t)(b * NH + h) * LSEQ + q) << 10) + k]);
        }
    }
}

// ---------------------------------------------------------------------------
// Kernel 2: gate_a_1[b,h,l] = sig(g0)*(sig(g1)*const[h]-1)+2
// ---------------------------------------------------------------------------
__global__ __launch_bounds__(256) void gate_kernel(
    const float* __restrict__ query, const float* __restrict__ gru_w,
    const float* __restrict__ gru_b, const float* __restrict__ gru_const,
    float* __restrict__ gates) {
    __shared__ float w[8 * DH];
    __shared__ float gb[8];
    if (threadIdx.x < 8) gb[threadIdx.x] = gru_b[threadIdx.x];
    for (int i = threadIdx.x; i < 8 * DH; i += 256) w[i] = gru_w[i];
    __syncthreads();

    int gid = blockIdx.x * 256 + threadIdx.x;   // (b*H+h)*L + l
    int l  = gid & (LSEQ - 1);
    int bh = gid >> 10;
    int b = bh / NH, h = bh % NH;

    const float* qp = query + (size_t)(b * LSEQ + l) * EMB + h * DH;
    float s[8];
#pragma unroll
    for (int j = 0; j < 8; ++j) s[j] = gb[j];
    for (int d = 0; d < DH; ++d) {
        float qv = qp[d];
#pragma unroll
        for (int j = 0; j < 8; ++j) s[j] = fmaf(qv, w[j * DH + d], s[j]);
    }
    float g0 = s[0] + s[1] + s[2] + s[3];
    float g1 = s[4] + s[5] + s[6] + s[7];
    g0 = 1.0f / (1.0f + __expf(-g0));
    g1 = 1.0f / (1.0f + __expf(-g1));
    gates[gid] = g0 * (g1 * gru_const[h] - 1.0f) + 2.0f;
}

// ---------------------------------------------------------------------------
// Kernel 3/5: GEMM  C[M,N] = A[M,K] @ W[N,K]^T + bias[N]
// 64x128 block tile, 8 waves (4x2), double-buffered LDS fed by async copies:
// prefetch tile k+1 while WMMAs consume tile k.  M,N,K must divide tiles.
// ---------------------------------------------------------------------------
__global__ __launch_bounds__(256) void gemm_kernel(
    const float* __restrict__ A, const float* __restrict__ W,
    const float* __restrict__ bias, float* __restrict__ C,
    int M, int N, int K) {
    __shared__ float As[2 * 64 * AS_STRIDE];
    __shared__ float Ws[2 * 128 * AS_STRIDE];

    const int m0 = blockIdx.y * 64;
    const int n0 = blockIdx.x * 128;
    const int lane = threadIdx.x & 31;
    const int wave = threadIdx.x >> 5;
    const int wm = wave & 3;
    const int wn = wave >> 2;
    const int nn = lane & 15;
    const int half = lane >> 4;

    v8f acc[4];
#pragma unroll
    for (int t = 0; t < 4; ++t) acc[t] = (v8f){0, 0, 0, 0, 0, 0, 0, 0};

    auto issue_tiles = [&](int k0, int bi) {
        float* Ab = &As[bi * 64 * AS_STRIDE];
        float* Wb = &Ws[bi * 128 * AS_STRIDE];
#pragma unroll
        for (int i = 0; i < 2; ++i) {            // A tile: 64x32 = 512 chunks
            int idx = threadIdx.x + i * 256;
            int r = idx >> 3, c4 = (idx & 7) * 4;
            async_copy16(&A[(size_t)(m0 + r) * K + k0 + c4],
                         &Ab[r * AS_STRIDE + c4]);
        }
#pragma unroll
        for (int i = 0; i < 4; ++i) {            // W tile: 128x32 = 1024 chunks
            int idx = threadIdx.x + i * 256;
            int r = idx >> 3, c4 = (idx & 7) * 4;
            async_copy16(&W[(size_t)(n0 + r) * K + k0 + c4],
                         &Wb[r * AS_STRIDE + c4]);
        }
    };

    issue_tiles(0, 0);
    wait_async0();
    __syncthreads();

    int buf = 0;
    for (int k0 = 0; k0 < K; k0 += 32, buf ^= 1) {
        if (k0 + 32 < K) issue_tiles(k0 + 32, buf ^ 1);
        const float* Ab = &As[buf * 64 * AS_STRIDE];
        const float* Wb = &Ws[buf * 128 * AS_STRIDE];
#pragma unroll
        for (int kk = 0; kk < 32; kk += 4) {
            const int c = kk + 2 * half;
            v2f af;
            af.x = Ab[(wm * 16 + nn) * AS_STRIDE + c];
            af.y = Ab[(wm * 16 + nn) * AS_STRIDE + c + 1];
#pragma unroll
            for (int t = 0; t < 4; ++t) {
                const int nl = wn * 64 + t * 16 + nn;
                v2f bf;
                bf.x = Wb[nl * AS_STRIDE + c];
                bf.y = Wb[nl * AS_STRIDE + c + 1];
                acc[t] = wmma_f32(af, bf, acc[t]);
            }
        }
        wait_async0();
        __syncthreads();
    }
#pragma unroll
    for (int t = 0; t < 4; ++t) {
        const int col = n0 + wn * 64 + t * 16 + nn;
        const float bv = bias[col];
#pragma unroll
        for (int r = 0; r < 8; ++r) {
            const int row = m0 + wm * 16 + r + 8 * half;
            C[(size_t)row * N + col] = acc[t][r] + bv;
        }
    }
}

// ---------------------------------------------------------------------------
// Kernel 4: fused flash attention per (b,h).  8 waves x 16 q-rows per block;
// K/V 16-key tiles double-buffered in LDS via async copies; online softmax in
// the WMMA C-fragment layout; P relayout through padded LDS; P.V via WMMA.
// ---------------------------------------------------------------------------
__global__ __launch_bounds__(256) void attn_kernel(
    const float* __restrict__ qkv, const float* __restrict__ gates,
    const float* __restrict__ bias_b, float* __restrict__ ctx) {
    const int lane = threadIdx.x & 31;
    const int wave = threadIdx.x >> 5;
    const int nn = lane & 15;
    const int half = lane >> 4;

    const int qchunks = LSEQ / 128;              // 8
    const int bh = blockIdx.x / qchunks;
    const int qc = blockIdx.x % qchunks;
    const int b = bh / NH, h = bh % NH;
    const int q0 = qc * 128 + wave * 16;

    __shared__ float Ks[2 * 16 * KS_STRIDE];
    __shared__ float Vs[2 * 16 * KS_STRIDE];
    __shared__ float Ps[8 * 16 * PS_STRIDE];

    // Q fragments, held in VGPRs for the whole pass: A[m][k], m = lane%16
    v2f aq[16];
    {
        const float* qb =
            qkv + (size_t)(b * LSEQ + q0 + nn) * QKV_STRIDE + h * DH;
#pragma unroll
        for (int t = 0; t < 16; ++t) {
            const int c = 4 * t + 2 * half;
            aq[t].x = qb[c];
            aq[t].y = qb[c + 1];
        }
    }
    float garow[8];
#pragma unroll
    for (int r = 0; r < 8; ++r)
        garow[r] = gates[(size_t)(b * NH + h) * LSEQ + q0 + r + 8 * half];

    float mprev[8], lsum[8];
#pragma unroll
    for (int r = 0; r < 8; ++r) { mprev[r] = -1e30f; lsum[r] = 0.f; }
    v8f O[4];
#pragma unroll
    for (int t = 0; t < 4; ++t) O[t] = (v8f){0, 0, 0, 0, 0, 0, 0, 0};

    const float scale = 0.125f;                  // 1/sqrt(64)
    const float* kvbase = qkv + (size_t)b * LSEQ * QKV_STRIDE;
    float* pw = Ps + wave * (16 * PS_STRIDE);

    auto issue_kv = [&](int kt, int bi) {
        float* Kb = &Ks[bi * 16 * KS_STRIDE];
        float* Vb = &Vs[bi * 16 * KS_STRIDE];
        int row = threadIdx.x >> 4;              // 16 rows
        int c4 = (threadIdx.x & 15) * 4;         // 16 chunks/row
        size_t gb = (size_t)(kt * 16 + row) * QKV_STRIDE + h * DH + c4;
        async_copy16(&kvbase[gb + EMB],     &Kb[row * KS_STRIDE + c4]);
        async_copy16(&kvbase[gb + 2 * EMB], &Vb[row * KS_STRIDE + c4]);
    };

    issue_kv(0, 0);
    wait_async0();
    __syncthreads();

    int buf = 0;
    for (int kt = 0; kt < LSEQ / 16; ++kt, buf ^= 1) {
        if (kt + 1 < LSEQ / 16) issue_kv(kt + 1, buf ^ 1);
        const float* Kb = &Ks[buf * 16 * KS_STRIDE];
        const float* Vb = &Vs[buf * 16 * KS_STRIDE];

        // ---- S = Q K^T : 16 WMMAs over head dim 64 ----
        v8f acc = (v8f){0, 0, 0, 0, 0, 0, 0, 0};
#pragma unroll
        for (int t = 0; t < 16; ++t) {
            const int c = 4 * t + 2 * half;
            v2f bf;                              // B[k][n] = K[key n][k]
            bf.x = Kb[nn * KS_STRIDE + c];
            bf.y = Kb[nn * KS_STRIDE + c + 1];
            acc = wmma_f32(aq[t], bf, acc);
        }

        // ---- bias + online softmax (rows r+8*half, cols = lane%16) ----
        const float* bp = bias_b +
            (((size_t)(b * NH + h) * LSEQ + q0 + 8 * half) << 10) +
            kt * 16 + nn;
        float p[8], alpha[8];
#pragma unroll
        for (int r = 0; r < 8; ++r) {
            float bias_v = __builtin_nontemporal_load(&bp[(size_t)r << 10]);
            float s = fmaf(garow[r], bias_v, acc[r] * scale);
            float mt = s;
            mt = fmaxf(mt, __shfl_xor(mt, 1, 32));
            mt = fmaxf(mt, __shfl_xor(mt, 2, 32));
            mt = fmaxf(mt, __shfl_xor(mt, 4, 32));
            mt = fmaxf(mt, __shfl_xor(mt, 8, 32));
            float mn = fmaxf(mprev[r], mt);
            alpha[r] = __expf(mprev[r] - mn);
            p[r] = __expf(s - mn);
            float rs = p[r];
            rs += __shfl_xor(rs, 1, 32);
            rs += __shfl_xor(rs, 2, 32);
            rs += __shfl_xor(rs, 4, 32);
            rs += __shfl_xor(rs, 8, 32);
            lsum[r] = lsum[r] * alpha[r] + rs;
            mprev[r] = mn;
        }
#pragma unroll
        for (int t = 0; t < 4; ++t)
#pragma unroll
            for (int r = 0; r < 8; ++r) O[t][r] *= alpha[r];

        // ---- relayout P (C-frag -> A-frag) through padded LDS ----
#pragma unroll
        for (int r = 0; r < 8; ++r)
            pw[(r + 8 * half) * PS_STRIDE + nn] = p[r];
        __syncthreads();

        // ---- O += P * V ----
#pragma unroll
        for (int ct = 0; ct < 4; ++ct) {
#pragma unroll
            for (int s4 = 0; s4 < 4; ++s4) {
                const int kk = 4 * s4 + 2 * half;
                v2f af, bf;
                af.x = pw[nn * PS_STRIDE + kk];
                af.y = pw[nn * PS_STRIDE + kk + 1];
                bf.x = Vb[kk * KS_STRIDE + ct * 16 + nn];
                bf.y = Vb[(kk + 1) * KS_STRIDE + ct * 16 + nn];
                O[ct] = wmma_f32(af, bf, O[ct]);
            }
        }
        wait_async0();
        __syncthreads();
    }

    // ---- normalize and write context (b, q, h*64 + d) ----
#pragma unroll
    for (int ct = 0; ct < 4; ++ct) {
#pragma unroll
        for (int r = 0; r < 8; ++r) {
            const int row = q0 + r + 8 * half;
            ctx[(size_t)(b * LSEQ + row) * EMB + h * DH + ct * 16 + nn] =
                O[ct][r] / lsum[r];
        }
    }
}

// ---------------------------------------------------------------------------
extern "C" void kernel_launch(void* const* d_in, const int* in_sizes, int n_in,
                              void* d_out, int out_size, void* d_ws,
                              size_t ws_size, hipStream_t stream) {
    const float* query     = (const float*)d_in[0];
    const float* in_w      = (const float*)d_in[1];
    const float* in_b      = (const float*)d_in[2];
    const float* out_w     = (const float*)d_in[3];
    const float* out_b     = (const float*)d_in[4];
    const float* rel_emb   = (const float*)d_in[5];
    const float* gru_w     = (const float*)d_in[6];
    const float* gru_b     = (const float*)d_in[7];
    const float* gru_const = (const float*)d_in[8];

    float* out1 = (float*)d_out;                          // (B,L,E)
    float* out2 = out1 + (size_t)BSZ * LSEQ * EMB;        // (B,H,L,L)

    float* ws    = (float*)d_ws;
    float* qkv   = ws;                                    // 8192*2304
    float* ctx   = qkv + (size_t)BSZ * LSEQ * QKV_STRIDE; // 8192*768
    float* gates = ctx + (size_t)BSZ * LSEQ * EMB;        // 98304

    const int M = BSZ * LSEQ;                             // 8192

    bias_kernel<<<(LSEQ * LSEQ) / 256, 256, 0, stream>>>(rel_emb, out2);
    gate_kernel<<<(BSZ * NH * LSEQ) / 256, 256, 0, stream>>>(
        query, gru_w, gru_b, gru_const, gates);
    gemm_kernel<<<dim3(QKV_STRIDE / 128, M / 64), 256, 0, stream>>>(
        query, in_w, in_b, qkv, M, QKV_STRIDE, EMB);
    attn_kernel<<<BSZ * NH * (LSEQ / 128), 256, 0, stream>>>(
        qkv, gates, out2, ctx);
    gemm_kernel<<<dim3(EMB / 128, M / 64), 256, 0, stream>>>(
        ctx, out_w, out_b, out1, M, EMB, EMB);
}